// SampledPathEnsemble_35424890257689
// MI455X (gfx1250) — compile-verified
//
#include <hip/hip_runtime.h>

// Problem constants (match reference)
#define N_FEATURE 256
#define DEPTH     12
#define N_TREE    512
#define N_BATCH   4096
#define N_OUTPUT  8
#define N_NODE    8191   // 2^(DEPTH+1) - 1

typedef __attribute__((ext_vector_type(2))) float v2f;
typedef __attribute__((ext_vector_type(8))) float v8f;

__global__ __launch_bounds__(N_TREE)
void SampledPathEnsemble_kernel(const float* __restrict__ x,
                                const int*   __restrict__ feature,
                                const float* __restrict__ threshold,
                                const int*   __restrict__ children_left,
                                const int*   __restrict__ children_right,
                                const float* __restrict__ value,
                                const float* __restrict__ scale,
                                const float* __restrict__ unif,
                                float*       __restrict__ out)
{
    __shared__ float s_x[N_FEATURE + 1];   // xp: x row + trailing zero for feat<0
    __shared__ float s_P[16][16];          // 16 wave-partials x 16 cols (8 used)

    const int t = threadIdx.x;             // tree id (0..511)
    const int b = blockIdx.x;              // batch id (0..4095)

    // Stage x[b,:] into LDS once; gathers per step then hit LDS, not HBM.
    if (t < N_FEATURE) s_x[t] = x[b * N_FEATURE + t];
    if (t == 0)        s_x[N_FEATURE] = 0.0f;
    __syncthreads();

    const float inv_s = 1.0f / scale[0];   // uniform -> scalar load
    const int   base  = t * N_NODE;

    int node = 0;
    #pragma unroll
    for (int s = 0; s < DEPTH; ++s) {
        const int   idx  = base + node;
        const int   feat = feature[idx];
        const float thr  = threshold[idx];
        const int   lch  = children_left[idx];
        const int   rch  = children_right[idx];
        const float u    = unif[(s * N_BATCH + b) * N_TREE + t];   // coalesced
        if (s + 1 < DEPTH)   // gfx1250: global_prefetch_b8 for next step's unif line
            __builtin_prefetch(&unif[((s + 1) * N_BATCH + b) * N_TREE + t], 0, 3);

        const int   fidx = (feat < 0) ? N_FEATURE : feat;
        const float xv   = s_x[fidx];
        const float d    = (thr - xv) * inv_s;
        // u < sigmoid(d)  <=>  u * (1 + exp(-d)) < 1   (no reciprocal needed)
        const float e    = __builtin_amdgcn_exp2f(d * -1.4426950408889634f);
        const bool  go_left = (u * (1.0f + e)) < 1.0f;
        if (lch > -1) node = go_left ? lch : rch;
    }

    // Gather leaf value row (32B, 32B-aligned)
    const float4* vp = (const float4*)(value + (size_t)(base + node) * N_OUTPUT);
    const float4 v0 = vp[0];
    const float4 v1 = vp[1];
    float acc[N_OUTPUT] = {v0.x, v0.y, v0.z, v0.w, v1.x, v1.y, v1.z, v1.w};

    // Wave32 butterfly reduction over the 32 trees in this wave.
    #pragma unroll
    for (int o = 0; o < N_OUTPUT; ++o) {
        #pragma unroll
        for (int off = 16; off > 0; off >>= 1)
            acc[o] += __shfl_xor(acc[o], off, 32);
    }

    const int wave = t >> 5;
    const int lane = t & 31;
    if (lane == 0) {
        #pragma unroll
        for (int o = 0; o < N_OUTPUT; ++o) s_P[wave][o] = acc[o];
        #pragma unroll
        for (int o = N_OUTPUT; o < 16; ++o) s_P[wave][o] = 0.0f;
    }
    __syncthreads();

    // Wave 0 finishes the cross-wave reduction with WMMA:
    // A = ones(16x4)  =>  D[m,n] = sum_k B[k,n]; accumulate 4 chunks of 4 rows
    // to get column sums of the full 16x16 partial tile. Since A is all-ones,
    // the K-row <-> (lane-half, vgpr) placement of B cannot affect the result.
    if (t < 32) {                           // whole wave 0 -> EXEC all ones
        v2f a; a.x = 1.0f; a.y = 1.0f;      // all-ones A operand
        v8f c = {};
        const int  nlo = lane & 15;         // column index
        const int  khi = (lane >= 16) ? 2 : 0;
        #pragma unroll
        for (int chunk = 0; chunk < 4; ++chunk) {
            const int k0 = 4 * chunk + khi;
            v2f bb;
            bb.x = s_P[k0 + 0][nlo];
            bb.y = s_P[k0 + 1][nlo];
            c = __builtin_amdgcn_wmma_f32_16x16x4_f32(
                    /*neg_a=*/false, a, /*neg_b=*/false, bb,
                    /*c_mod=*/(short)0, c, /*reuse_a=*/false, /*reuse_b=*/false);
        }
        // C/D layout: lane n (0..15), VGPR0 = D[0][n] = column-n sum.
        if (lane < N_OUTPUT)
            out[b * N_OUTPUT + lane] = c[0] * (1.0f / N_TREE);
    }
}

extern "C" void kernel_launch(void* const* d_in, const int* in_sizes, int n_in,
                              void* d_out, int out_size, void* d_ws, size_t ws_size,
                              hipStream_t stream) {
    const float* x              = (const float*)d_in[0];
    const int*   feature        = (const int*)  d_in[1];
    const float* threshold      = (const float*)d_in[2];
    const int*   children_left  = (const int*)  d_in[3];
    const int*   children_right = (const int*)  d_in[4];
    const float* value          = (const float*)d_in[5];
    const float* scale          = (const float*)d_in[6];
    const float* unif           = (const float*)d_in[7];
    float*       out            = (float*)d_out;

    SampledPathEnsemble_kernel<<<N_BATCH, N_TREE, 0, stream>>>(
        x, feature, threshold, children_left, children_right,
        value, scale, unif, out);
}